// TensorTrainGaussian2D_27960237097561
// MI455X (gfx1250) — compile-verified
//
#include <hip/hip_runtime.h>
#include <hip/hip_bf16.h>
#include <math.h>

// Tensor-train Gaussian 2D likelihood, K=128, N=4096.
// Strategy: pdf exponent is rank-3 in x ([x^2, x, 1]) -> evaluate 16x16 tiles
// of exponents with V_WMMA_F32_16X16X4_F32, exp2 on the TRANS pipe, weighted
// reductions on VALU. Softmax log-weights and 1/sigma^2 are folded into the
// WMMA A-matrix coefficients (log2 domain) by a one-shot prep kernel.

#define KDIM 128
#define NPTS 4096
#define HALF_LOG_2PI 0.9189385332046727f
#define LOG2E        1.4426950408889634f
#define EPS_F        2.220446049250313e-16f

typedef __attribute__((ext_vector_type(2))) float v2f;
typedef __attribute__((ext_vector_type(4))) float v4f;
typedef __attribute__((ext_vector_type(8))) float v8f;

// ---------------------------------------------------------------------------
// Prep: column softmaxes (axis 0) of Wk1k0 / Wk2k1, softmax of Wk0, and the
// per-(row,col) quadratic coefficients (log2 domain) packed as float4 so a
// WMMA A-fragment is a single b64 load per lane:
//   lanes 0-15  load (f0,f1) = (a*log2e, b*log2e)       [K-features 0,1]
//   lanes 16-31 load (f2, 0) = ((c+logw)*log2e, 0)      [K-features 2,3]
// F4: stage-2 table indexed [k2, k1] (w2 folded in).
// G4: stage-1 table indexed [k1, k0] (w1 folded in).
// ---------------------------------------------------------------------------
__global__ __launch_bounds__(128) void ttg_prep(
    const float* __restrict__ Wk0, const float* __restrict__ W1,
    const float* __restrict__ W2, const float* __restrict__ mu,
    const float* __restrict__ sigma,
    float* __restrict__ w0_out, v4f* __restrict__ F4, v4f* __restrict__ G4)
{
  const int t = threadIdx.x;  // column index 0..127

  // log-partition of column t for W1 and W2 (softmax along axis 0 = rows)
  float m1 = -INFINITY, m2 = -INFINITY;
  for (int r = 0; r < KDIM; ++r) {
    m1 = fmaxf(m1, W1[r * KDIM + t]);
    m2 = fmaxf(m2, W2[r * KDIM + t]);
  }
  float s1 = 0.f, s2 = 0.f;
  for (int r = 0; r < KDIM; ++r) {
    s1 += __expf(W1[r * KDIM + t] - m1);
    s2 += __expf(W2[r * KDIM + t] - m2);
  }
  const float logZ1 = m1 + __logf(s1);
  const float logZ2 = m2 + __logf(s2);

  // w0 softmax (each thread redundantly scans 128 values; trivial)
  float m0 = -INFINITY;
  for (int k = 0; k < KDIM; ++k) m0 = fmaxf(m0, Wk0[k]);
  float s0 = 0.f;
  for (int k = 0; k < KDIM; ++k) s0 += __expf(Wk0[k] - m0);
  w0_out[t] = __expf(Wk0[t] - m0) / s0;

  // coefficient tables
  for (int r = 0; r < KDIM; ++r) {
    const int e = r * KDIM + t;
    const float sg = sigma[e], m = mu[e];
    const float a = -0.5f / (sg * sg);
    const float b = -2.f * a * m;
    const float c = a * m * m - __logf(sg) - HALF_LOG_2PI;
    v4f f2;
    f2.x = a * LOG2E;
    f2.y = b * LOG2E;
    f2.z = (c + W2[e] - logZ2) * LOG2E;
    f2.w = 0.f;
    F4[e] = f2;
    v4f f1;
    f1.x = a * LOG2E;
    f1.y = b * LOG2E;
    f1.z = (c + W1[e] - logZ1) * LOG2E;
    f1.w = 0.f;
    G4[e] = f1;
  }
}

// ---------------------------------------------------------------------------
// Main: one block (4 waves) per 16-sample tile.
// Stage 2: v[l,n] = sum_k exp2(Q2[k,l,n])         (w2 folded into Q2)
// Stage 1: L[n]  = sum_k w0[k] sum_l exp2(Q1[l,k,n]) * v[l,n]
// Q tiles (16 rows x 16 n) come from V_WMMA_F32_16X16X4_F32.
// ---------------------------------------------------------------------------
__global__ __launch_bounds__(128) void ttg_main(
    const float* __restrict__ X, const float* __restrict__ w0,
    const float* __restrict__ F4, const float* __restrict__ G4,
    float* __restrict__ out)
{
  __shared__ float lds_v[KDIM * 17];  // v[l][n], row stride 17 (conflict pad)
  __shared__ float lds_L[4 * 16];     // per-wave partial L

  const int tid  = threadIdx.x;
  const int wave = tid >> 5;
  const int lane = tid & 31;
  const int half = lane >> 4;  // 0: K-features {0,1} / rows M 0-7 of D
  const int ln   = lane & 15;  // sample index within the tile
  const int nbase = blockIdx.x * 16;

  const float x1 = X[(nbase + ln) * 2 + 0];
  const float x2 = X[(nbase + ln) * 2 + 1];

  // B fragments: 4x16 feature matrix [x^2, x, 1, 0] striped per ISA layout:
  // lanes 0-15 hold K-rows 0,1 ; lanes 16-31 hold K-rows 2,3.
  v2f bf1, bf2;
  bf1.x = half ? 1.0f : x1 * x1;
  bf1.y = half ? 0.0f : x1;
  bf2.x = half ? 1.0f : x2 * x2;
  bf2.y = half ? 0.0f : x2;

  const v2f* __restrict__ F2 = (const v2f*)F4;
  const v2f* __restrict__ G2 = (const v2f*)G4;

  // -------------------- stage 2: v[l][n] --------------------
  for (int l = wave; l < KDIM; l += 4) {
    float acc = 0.f;
    for (int kt = 0; kt < 8; ++kt) {
      const int row = kt * 16 + ln;                 // k row for this lane
      const v2f a = F2[(row * KDIM + l) * 2 + half];
      v8f d = {0.f, 0.f, 0.f, 0.f, 0.f, 0.f, 0.f, 0.f};
      d = __builtin_amdgcn_wmma_f32_16x16x4_f32(
          /*neg_a=*/false, a, /*neg_b=*/false, bf2,
          /*c_mod=*/(short)0, d, /*reuse_a=*/false, /*reuse_b=*/false);
#pragma unroll
      for (int i = 0; i < 8; ++i) acc += __builtin_amdgcn_exp2f(d[i]);
    }
    acc += __shfl_xor(acc, 16, 32);  // combine row-halves (wave32)
    if (half == 0) lds_v[l * 17 + ln] = acc;
  }
  __syncthreads();

  // -------------------- stage 1: L[n] --------------------
  float Lacc = 0.f;
  for (int k = wave; k < KDIM; k += 4) {
    const float w0k = w0[k];
    float uacc = 0.f;
    for (int lt = 0; lt < 8; ++lt) {
      const int row = lt * 16 + ln;                 // l row for this lane
      const v2f a = G2[(row * KDIM + k) * 2 + half];
      v8f d = {0.f, 0.f, 0.f, 0.f, 0.f, 0.f, 0.f, 0.f};
      d = __builtin_amdgcn_wmma_f32_16x16x4_f32(
          false, a, false, bf1, (short)0, d, false, false);
      const int rbase = lt * 16 + half * 8;         // D rows: lanes>=16 hold M+8
#pragma unroll
      for (int i = 0; i < 8; ++i)
        uacc += __builtin_amdgcn_exp2f(d[i]) * lds_v[(rbase + i) * 17 + ln];
    }
    uacc += __shfl_xor(uacc, 16, 32);
    Lacc += w0k * uacc;
  }
  if (half == 0) lds_L[wave * 16 + ln] = Lacc;
  __syncthreads();

  if (tid < 16) {
    const float L = lds_L[tid] + lds_L[16 + tid] + lds_L[32 + tid] + lds_L[48 + tid];
    out[nbase + tid] = __logf(L + EPS_F);
  }
}

// ---------------------------------------------------------------------------
extern "C" void kernel_launch(void* const* d_in, const int* in_sizes, int n_in,
                              void* d_out, int out_size, void* d_ws, size_t ws_size,
                              hipStream_t stream) {
  const float* X     = (const float*)d_in[0];  // (N,2)
  const float* Wk0   = (const float*)d_in[1];  // (K,)
  const float* Wk1k0 = (const float*)d_in[2];  // (K,K)
  const float* Wk2k1 = (const float*)d_in[3];  // (K,K)
  const float* mu    = (const float*)d_in[4];  // (K,K)
  const float* sigma = (const float*)d_in[5];  // (K,K)
  float* out = (float*)d_out;

  // Workspace layout (needs ~525 KB): w0 | F4 (stage2 coeffs) | G4 (stage1)
  float* w0 = (float*)d_ws;
  v4f* F4 = (v4f*)((char*)d_ws + 512);
  v4f* G4 = (v4f*)((char*)d_ws + 512 + (size_t)KDIM * KDIM * sizeof(v4f));

  ttg_prep<<<1, 128, 0, stream>>>(Wk0, Wk1k0, Wk2k1, mu, sigma, w0, F4, G4);
  ttg_main<<<NPTS / 16, 128, 0, stream>>>(X, w0, (const float*)F4,
                                          (const float*)G4, out);
}